// AdaptiveBiasReflectiveLayerV7_26869315404620
// MI455X (gfx1250) — compile-verified
//
#include <hip/hip_runtime.h>
#include <hip/hip_bf16.h>
#include <math.h>

// Shapes fixed by the reference
#define HDIM 4096
#define RDIM 256
#define KC   32      // K-chunk staged in LDS
#define MROWS 64     // rows of BS per workgroup (4 m-tiles x 2 n-halves over 8 waves)
#define EPSV 1e-6f

typedef float v2f __attribute__((ext_vector_type(2)));
typedef float v8f __attribute__((ext_vector_type(8)));
typedef int   v4i __attribute__((vector_size(16)));   // matches builtin param type

// ---- async global->LDS copy (CDNA5) with portable fallback ----------------
#if __has_builtin(__builtin_amdgcn_global_load_async_to_lds_b128)
#define HAVE_ASYNC_LDS 1
#define ASYNC_CP16(gsrc, ldst)                                                 \
    __builtin_amdgcn_global_load_async_to_lds_b128(                            \
        (__attribute__((address_space(1))) v4i*)(gsrc),                        \
        (__attribute__((address_space(3))) v4i*)(ldst), 0, 0)
#if __has_builtin(__builtin_amdgcn_s_wait_asynccnt)
#define ASYNC_WAIT(n) __builtin_amdgcn_s_wait_asynccnt(n)
#else
#define ASYNC_WAIT(n) asm volatile("s_wait_asynccnt %0" ::"i"(n) : "memory")
#endif
#else
#define HAVE_ASYNC_LDS 0
#define ASYNC_CP16(gsrc, ldst)                                                 \
    (*(float4*)(ldst) = *(const float4*)(gsrc))
#define ASYNC_WAIT(n) ((void)0)
#endif

// ---------------------------------------------------------------- init
__global__ __launch_bounds__(512)
void init_ws_kernel(float* ws) {
    // zero sumP[256] and sumP2[256]
    ws[threadIdx.x] = 0.0f;
}

// ------------------------------------------- kernel 1: P = x @ proj^T stats
// Per-column sum / sum-of-squares of P over all BS rows via f32 WMMA, with
// double-buffered async global->LDS staging.
__global__ __launch_bounds__(256)
void gemm_stats_kernel(const float* __restrict__ x,
                       const float* __restrict__ proj,
                       float* __restrict__ gsum,
                       float* __restrict__ gsumsq) {
    __shared__ float xs[2][MROWS * KC];   // 2 x 8 KB
    __shared__ float ps[2][RDIM * KC];    // 2 x 32 KB
    __shared__ float colsum[RDIM];
    __shared__ float colsq[RDIM];

    const int tid  = threadIdx.x;
    const int lane = tid & 31;
    const int wave = tid >> 5;          // 0..7
    const int mt   = wave >> 1;         // 0..3 : M-tile
    const int nh   = wave & 1;          // 0..1 : N half (128 cols each)
    const int m0   = mt * 16;
    const int row0 = blockIdx.x * MROWS;
    const int laneM = lane & 15;
    const int koff  = (lane & 16) ? 2 : 0;   // lanes 16-31 hold K=2,3

    if (tid < RDIM) { colsum[tid] = 0.0f; colsq[tid] = 0.0f; }

    v8f acc[8];
    #pragma unroll
    for (int nt = 0; nt < 8; ++nt)
        #pragma unroll
        for (int i = 0; i < 8; ++i) acc[nt][i] = 0.0f;

    // Stage one K-chunk (x tile 64x32, proj tile 256x32) into buffer b.
    // 2 + 8 = 10 async b128 copies per thread.
    auto stage = [&](int k0, int b) {
        #pragma unroll
        for (int j = 0; j < 2; ++j) {           // x: 512 float4
            int f = tid + j * 256;
            int r = f >> 3, c = (f & 7) * 4;
            ASYNC_CP16(x + (size_t)(row0 + r) * HDIM + k0 + c,
                       &xs[b][r * KC + c]);
        }
        #pragma unroll
        for (int j = 0; j < 8; ++j) {           // proj: 2048 float4
            int f = tid + j * 256;
            int r = f >> 3, c = (f & 7) * 4;
            ASYNC_CP16(proj + (size_t)r * HDIM + k0 + c,
                       &ps[b][r * KC + c]);
        }
    };

    stage(0, 0);
    int cur = 0;
    for (int k0 = 0; k0 < HDIM; k0 += KC) {
        if (k0 + KC < HDIM) {
            stage(k0 + KC, cur ^ 1);   // overlap next chunk's copies
            ASYNC_WAIT(10);            // wait for current chunk only
        } else {
            ASYNC_WAIT(0);
        }
        __syncthreads();               // all waves' copies for 'cur' landed

        const float* xb = &xs[cur][(m0 + laneM) * KC + koff];
        const float* pb = &ps[cur][(nh * 128 + laneM) * KC + koff];
        for (int k = 0; k < KC; k += 4) {
            v2f a = *(const v2f*)(xb + k);       // A: rows m0..m0+15, K=k..k+3
            #pragma unroll
            for (int nt = 0; nt < 8; ++nt) {
                // B[k][n] = proj[n][k]; n = nh*128 + nt*16 + laneM
                v2f b = *(const v2f*)(pb + nt * 16 * KC + k);
                acc[nt] = __builtin_amdgcn_wmma_f32_16x16x4_f32(
                    false, a, false, b, (short)0, acc[nt], false, false);
            }
        }
        __syncthreads();               // buffer 'cur' free for restaging
        cur ^= 1;
    }

    // VGPR i of tile nt holds P[m0+i (+8 if lane>=16)][nh*128 + nt*16 + laneM]
    #pragma unroll
    for (int nt = 0; nt < 8; ++nt) {
        float s = 0.0f, s2 = 0.0f;
        #pragma unroll
        for (int i = 0; i < 8; ++i) { float v = acc[nt][i]; s += v; s2 += v * v; }
        int col = nh * 128 + nt * 16 + laneM;
        atomicAdd(&colsum[col], s);
        atomicAdd(&colsq[col], s2);
    }
    __syncthreads();
    if (tid < RDIM) {
        atomicAdd(&gsum[tid],   colsum[tid]);
        atomicAdd(&gsumsq[tid], colsq[tid]);
    }
}

// ------------------------------------------- kernel 2: analytic scale loop
__device__ __forceinline__
float block_sum256(float v, float* red, int tid) {
    __syncthreads();
    red[tid] = v;
    __syncthreads();
    for (int off = 128; off > 0; off >>= 1) {
        if (tid < off) red[tid] += red[tid + off];
        __syncthreads();
    }
    float r = red[0];
    __syncthreads();
    return r;
}

__global__ __launch_bounds__(256)
void correction_kernel(const float* __restrict__ gsum,
                       const float* __restrict__ gsumsq,
                       const float* __restrict__ proj,
                       const float* __restrict__ proj_bias,
                       const float* __restrict__ proj_weights,
                       const float* __restrict__ ref_mu,
                       const float* __restrict__ ref_sigma,
                       float* __restrict__ c_out,
                       int NBS) {
    __shared__ float dw[RDIM];
    __shared__ float qs[RDIM];
    __shared__ float red[RDIM];
    __shared__ float corr[HDIM];   // 16 KB
    __shared__ float cl[HDIM];     // 16 KB

    const int tid = threadIdx.x;   // == r index
    const float invN = 1.0f / (float)NBS;

    const float muP  = gsum[tid] * invN;
    const float ex2  = gsumsq[tid] * invN;
    const float stdP = sqrtf(fmaxf(ex2 - muP * muP, 0.0f));
    const float biasr = proj_bias[tid];
    const float rmu   = ref_mu[tid];
    const float rsig  = ref_sigma[tid];

    qs[tid] = 0.0f;
    #pragma unroll
    for (int j = 0; j < HDIM / 256; ++j) cl[tid + j * 256] = 0.0f;

    const float scales[3] = {1.0f, 0.5f, 0.1f};
    const float threshold = 0.1f * (1.0f + 1.0f);   // KL_THRESHOLD*(1+VARIANCE_EMA)

    for (int it = 0; it < 3; ++it) {
        const float scale = scales[it];
        __syncthreads();
        const float w  = 1.0f / (1.0f + expf(-proj_weights[it * RDIM + tid]));
        const float sw = scale * w;
        const float mu = sw * (muP + qs[tid]) + biasr;
        const float sigma = fmaxf(sw * stdP, EPSV);
        const float inv2s2 = 0.5f / (sigma * sigma);
        const float lg = logf(sigma / rsig + EPSV);
        float d0 = rmu - mu;
        float klr = lg + (rsig * rsig + d0 * d0) * inv2s2 - 0.5f;
        float delta = mu - rmu;
        dw[tid] = delta * w;

        float kl  = block_sum256(klr, red, tid) * (1.0f / RDIM);
        float mad = block_sum256(fabsf(delta), red, tid) * (1.0f / RDIM);
        float alpha = 0.01f * fminf(fmaxf(mad, 0.05f), 10.0f);
        float coef = alpha * scale;

        // correction[h] = coef * sum_r dw[r]*proj[r][h]
        float tacc[HDIM / 256];
        #pragma unroll
        for (int j = 0; j < HDIM / 256; ++j) tacc[j] = 0.0f;
        for (int rr = 0; rr < RDIM; ++rr) {
            float d = dw[rr];
            const float* pr = proj + (size_t)rr * HDIM + tid;
            #pragma unroll
            for (int j = 0; j < HDIM / 256; ++j) tacc[j] += d * pr[j * 256];
        }
        #pragma unroll
        for (int j = 0; j < HDIM / 256; ++j) corr[tid + j * 256] = coef * tacc[j];
        __syncthreads();

        // q update: qd[r] = corr . proj[r][:]
        float s = 0.0f;
        const float* pr = proj + (size_t)tid * HDIM;
        for (int h = 0; h < HDIM; ++h) s += corr[h] * pr[h];
        float qpost = qs[tid] + s;

        float mu2 = sw * (muP + qpost) + biasr;   // sigma unchanged by a shift
        float d2 = rmu - mu2;
        float klr2 = lg + (rsig * rsig + d2 * d2) * inv2s2 - 0.5f;
        float kl_post = block_sum256(klr2, red, tid) * (1.0f / RDIM);

        bool apply = (kl > threshold) && (kl_post < kl);
        if (apply) {
            qs[tid] = qpost;
            #pragma unroll
            for (int j = 0; j < HDIM / 256; ++j)
                cl[tid + j * 256] += corr[tid + j * 256];
        }
    }
    __syncthreads();
    #pragma unroll
    for (int j = 0; j < HDIM / 256; ++j) c_out[tid + j * 256] = cl[tid + j * 256];
}

// ------------------------------------------- kernel 3: row norm (ddof=1)
__global__ __launch_bounds__(256)
void norm_kernel(const float* __restrict__ x,
                 const float* __restrict__ cvec,
                 const float* __restrict__ gamma,
                 const float* __restrict__ beta,
                 float* __restrict__ out) {
    __shared__ float red[256];
    const int tid = threadIdx.x;
    const size_t row = blockIdx.x;

    const float4* xr = (const float4*)(x + row * HDIM);
    const float4* c4 = (const float4*)cvec;

    float4 v[4];
    float s = 0.0f, s2 = 0.0f;
    #pragma unroll
    for (int j = 0; j < 4; ++j) {
        int i4 = tid + j * 256;
        float4 xv = xr[i4];
        float4 cv = c4[i4];
        float4 t = make_float4(xv.x + cv.x, xv.y + cv.y, xv.z + cv.z, xv.w + cv.w);
        v[j] = t;
        s  += t.x + t.y + t.z + t.w;
        s2 += t.x * t.x + t.y * t.y + t.z * t.z + t.w * t.w;
    }
    // reduce sum
    red[tid] = s; __syncthreads();
    for (int off = 128; off > 0; off >>= 1) {
        if (tid < off) red[tid] += red[tid + off];
        __syncthreads();
    }
    float S = red[0]; __syncthreads();
    // reduce sumsq
    red[tid] = s2; __syncthreads();
    for (int off = 128; off > 0; off >>= 1) {
        if (tid < off) red[tid] += red[tid + off];
        __syncthreads();
    }
    float S2 = red[0];

    const float mean = S * (1.0f / HDIM);
    float var = (S2 - (float)HDIM * mean * mean) * (1.0f / (HDIM - 1));
    float sd = sqrtf(fmaxf(var, 0.0f));
    sd = fmaxf(sd, 10.0f * EPSV);            // max(std, EPS*10)
    const float inv = 1.0f / (sd + EPSV);

    float4* o = (float4*)(out + row * HDIM);
    const float4* g4 = (const float4*)gamma;
    const float4* b4 = (const float4*)beta;
    #pragma unroll
    for (int j = 0; j < 4; ++j) {
        int i4 = tid + j * 256;
        float4 g = g4[i4], b = b4[i4], t = v[j];
        float4 r = make_float4((t.x - mean) * inv * g.x + b.x,
                               (t.y - mean) * inv * g.y + b.y,
                               (t.z - mean) * inv * g.z + b.z,
                               (t.w - mean) * inv * g.w + b.w);
        o[i4] = r;
    }
}

// ---------------------------------------------------------------- launch
extern "C" void kernel_launch(void* const* d_in, const int* in_sizes, int n_in,
                              void* d_out, int out_size, void* d_ws, size_t ws_size,
                              hipStream_t stream) {
    const float* x            = (const float*)d_in[0];
    const float* gamma        = (const float*)d_in[1];
    const float* beta         = (const float*)d_in[2];
    const float* proj         = (const float*)d_in[3];
    const float* proj_bias    = (const float*)d_in[4];
    const float* proj_weights = (const float*)d_in[5];
    const float* ref_mu       = (const float*)d_in[6];
    const float* ref_sigma    = (const float*)d_in[7];
    float* out = (float*)d_out;
    float* wsf = (float*)d_ws;

    const int BS = in_sizes[0] / HDIM;   // 8192

    float* gsum   = wsf;           // [256]
    float* gsumsq = wsf + RDIM;    // [256]
    float* cvec   = wsf + 2 * RDIM;// [4096]

    init_ws_kernel<<<1, 512, 0, stream>>>(wsf);
    gemm_stats_kernel<<<BS / MROWS, 256, 0, stream>>>(x, proj, gsum, gsumsq);
    correction_kernel<<<1, 256, 0, stream>>>(gsum, gsumsq, proj, proj_bias,
                                             proj_weights, ref_mu, ref_sigma,
                                             cvec, BS);
    norm_kernel<<<BS, 256, 0, stream>>>(x, cvec, gamma, beta, out);
}